// CombinedLoss_15547781611525
// MI455X (gfx1250) — compile-verified
//
#include <hip/hip_runtime.h>
#include <hip/hip_bf16.h>

typedef float v2f __attribute__((ext_vector_type(2)));
typedef float v8f __attribute__((ext_vector_type(8)));

#define HW          262144      // 512*512
#define NCLS        6
#define NIMG        16
#define BLK_PER_IMG 16
#define THREADS     256
#define NSTAT       19          // [0]=ce, [1..6]=TP, [7..12]=Psum, [13..18]=Tsum
#define L2E         1.44269504088896340736f
#define LN2         0.69314718055994530942f

__device__ __forceinline__ float f4get(const float4& v, int p) {
    switch (p) { case 0: return v.x; case 1: return v.y; case 2: return v.z; default: return v.w; }
}

// ---------------------------------------------------------------------------
// Kernel 1: streaming softmax-statistics pass.
// Grid: 256 blocks (16 per image) x 256 threads; each thread handles 64 pixels
// as 16 iterations of float4 (wave reads 512B contiguous per class plane).
// Block-level reduction of the 19 per-thread stats uses V_WMMA_F32_16X16X4_F32
// with an all-ones A matrix: D[i][j] = sum_k B[k][j], i.e. 4-row column sums
// per instruction, accumulated over 64 pre-folded rows. f32 throughout.
// ---------------------------------------------------------------------------
__global__ __launch_bounds__(THREADS) void loss_stats_kernel(
    const float* __restrict__ logits,
    const int*   __restrict__ tgt2,      // int64 targets viewed as int pairs (low word valid)
    float*       __restrict__ part)      // [gridDim.x][NSTAT] block partials
{
    const int tid = threadIdx.x;
    const int img = blockIdx.x >> 4;     // /BLK_PER_IMG
    const int blk = blockIdx.x & 15;

    const float* base  = logits + (size_t)img * ((size_t)NCLS * HW);
    const int*   tbase = tgt2   + 2 * ((size_t)img * HW);

    float ce = 0.f;
    float tp[NCLS], ps[NCLS], ts[NCLS];
    #pragma unroll
    for (int c = 0; c < NCLS; ++c) { tp[c] = 0.f; ps[c] = 0.f; ts[c] = 0.f; }

    #pragma unroll 4
    for (int it = 0; it < 16; ++it) {
        const int pix = blk * 16384 + it * 1024 + tid * 4;

        float4 v[NCLS];
        #pragma unroll
        for (int c = 0; c < NCLS; ++c)
            v[c] = *(const float4*)(base + (size_t)c * HW + pix);

        const int4 ta = *(const int4*)(tbase + 2 * pix);       // t0,t1 (lo/hi pairs)
        const int4 tb = *(const int4*)(tbase + 2 * pix + 4);   // t2,t3
        const int tv4[4] = { ta.x, ta.z, tb.x, tb.z };

        #pragma unroll
        for (int p = 0; p < 4; ++p) {
            float x[NCLS];
            #pragma unroll
            for (int c = 0; c < NCLS; ++c) x[c] = f4get(v[c], p);

            float m = fmaxf(fmaxf(fmaxf(x[0], x[1]), fmaxf(x[2], x[3])), fmaxf(x[4], x[5]));
            float e[NCLS];
            #pragma unroll
            for (int c = 0; c < NCLS; ++c) e[c] = __builtin_amdgcn_exp2f((x[c] - m) * L2E);
            float s   = ((e[0] + e[1]) + (e[2] + e[3])) + (e[4] + e[5]);
            float lse = m + __builtin_amdgcn_logf(s) * LN2;    // v_log_f32 is log2
            float rs  = __builtin_amdgcn_rcpf(s);

            const int t = tv4[p];
            float xt = x[0], et = e[0];
            #pragma unroll
            for (int c = 1; c < NCLS; ++c) {
                bool hit = (t == c);
                xt = hit ? x[c] : xt;
                et = hit ? e[c] : et;
            }
            ce += (lse - xt);                 // -log p_t
            float pt = et * rs;               // prob at target class

            #pragma unroll
            for (int c = 0; c < NCLS; ++c) {
                ps[c] = fmaf(e[c], rs, ps[c]);          // sum of probs
                float oh = (t == c) ? 1.f : 0.f;
                ts[c] += oh;                            // one-hot counts
                tp[c]  = fmaf(oh, pt, tp[c]);           // TP = sum prob_t over t==c
            }
        }
    }

    // ---- stage per-thread stats into LDS (padded stride 33 -> conflict-free)
    __shared__ float red[THREADS][33];
    red[tid][0] = ce;
    #pragma unroll
    for (int c = 0; c < NCLS; ++c) {
        red[tid][1 + c]  = tp[c];
        red[tid][7 + c]  = ps[c];
        red[tid][13 + c] = ts[c];
    }
    #pragma unroll
    for (int s = NSTAT; s < 32; ++s) red[tid][s] = 0.f;
    __syncthreads();

    // ---- fold 256 rows -> 64 rows
    if (tid < 128) {
        #pragma unroll
        for (int s = 0; s < 32; ++s) red[tid][s] += red[tid + 128][s];
    }
    __syncthreads();
    if (tid < 64) {
        #pragma unroll
        for (int s = 0; s < 32; ++s) red[tid][s] += red[tid + 64][s];
    }
    __syncthreads();

    // ---- wave 0: WMMA column-sum reduction of 64 rows x 32 stats.
    // A = ones (16x4), B = 4x32-row chunk of stats; D[0][j] = total of stat j.
    // B f32 4x16 lane layout (mirrors documented A 16x4 K-split):
    //   VGPR0: lanes0-15 K=0, lanes16-31 K=2 ; VGPR1: K=1 / K=3.
    if (tid < 32) {
        const int lane = tid;
        const int col  = lane & 15;
        const int krow = (lane >> 4) * 2;

        v2f a; a[0] = 1.f; a[1] = 1.f;
        v8f acc0 = {}; v8f acc1 = {};

        #pragma unroll
        for (int k0 = 0; k0 < 64; k0 += 4) {
            v2f b0, b1;
            b0[0] = red[k0 + krow    ][col];
            b0[1] = red[k0 + krow + 1][col];
            b1[0] = red[k0 + krow    ][16 + col];
            b1[1] = red[k0 + krow + 1][16 + col];
            acc0 = __builtin_amdgcn_wmma_f32_16x16x4_f32(
                       false, a, false, b0, (short)0, acc0, false, false);
            acc1 = __builtin_amdgcn_wmma_f32_16x16x4_f32(
                       false, a, false, b1, (short)0, acc1, false, false);
        }

        float* dst = part + (size_t)blockIdx.x * NSTAT;
        if (lane < 16) dst[lane]      = acc0[0];  // D[0][col] = stats 0..15
        if (lane < 3)  dst[16 + lane] = acc1[0];  // stats 16..18
    }
}

// ---------------------------------------------------------------------------
// Kernel 2: deterministic fixed-order finalize over 256 block partials.
// ---------------------------------------------------------------------------
__global__ __launch_bounds__(128) void loss_finalize_kernel(
    const float* __restrict__ part, float* __restrict__ out)
{
    __shared__ float sce[128], sdice[128], sft[128];
    const int i = threadIdx.x;

    float ce = 0.f;
    for (int j = i; j < NIMG * BLK_PER_IMG; j += 128) ce += part[(size_t)j * NSTAT + 0];

    float dterm = 0.f, fterm = 0.f;
    if (i < NIMG * NCLS) {
        const int b = i / NCLS, c = i % NCLS;
        float tp = 0.f, psum = 0.f, tsum = 0.f;
        for (int k = 0; k < BLK_PER_IMG; ++k) {
            const float* p = part + (size_t)(b * BLK_PER_IMG + k) * NSTAT;
            tp += p[1 + c]; psum += p[7 + c]; tsum += p[13 + c];
        }
        float dice = (2.f * tp + 1e-8f) / (psum + tsum + 1e-8f);
        dterm = 1.f - dice;
        float fp = psum - tp, fn = tsum - tp;
        float tv = (tp + 1e-6f) / (tp + 0.7f * fn + 0.3f * fp + 1e-6f);
        fterm = __powf(fmaxf(1.f - tv, 0.f), 1.33f);
    }

    sce[i] = ce; sdice[i] = dterm; sft[i] = fterm;
    __syncthreads();
    for (int off = 64; off > 0; off >>= 1) {
        if (i < off) { sce[i] += sce[i + off]; sdice[i] += sdice[i + off]; sft[i] += sft[i + off]; }
        __syncthreads();
    }
    if (i == 0) {
        const float n_px = 4194304.f;   // 16*512*512
        const float n_bc = 96.f;        // 16*6
        out[0] = 0.4f * (sce[0] / n_px) + 0.4f * (sdice[0] / n_bc) + 0.2f * (sft[0] / n_bc);
    }
}

extern "C" void kernel_launch(void* const* d_in, const int* in_sizes, int n_in,
                              void* d_out, int out_size, void* d_ws, size_t ws_size,
                              hipStream_t stream) {
    const float* logits = (const float*)d_in[0];
    const int*   tgt2   = (const int*)d_in[1];   // int64 targets, low words
    float* out  = (float*)d_out;
    float* part = (float*)d_ws;                  // 256*19 floats = 19.5 KB

    loss_stats_kernel<<<NIMG * BLK_PER_IMG, THREADS, 0, stream>>>(logits, tgt2, part);
    loss_finalize_kernel<<<1, 128, 0, stream>>>(part, out);
}